// PhysicsMambaBlock_81862076662145
// MI455X (gfx1250) — compile-verified
//
#include <hip/hip_runtime.h>
#include <hip/hip_bf16.h>
#include <math.h>

#define D_MODEL 1024
#define D_INNER 2048
#define D_STATE 16
#define D_CONV  4
#define DT_RANK 64
#define BATCH   2
#define SEQ     2048
#define BL      (BATCH * SEQ)       // 4096 tokens
#define NCHUNK  16
#define CHLEN   (SEQ / NCHUNK)      // 128 serial steps per chunk

typedef float v2f __attribute__((ext_vector_type(2)));
typedef float v8f __attribute__((ext_vector_type(8)));

// ===========================================================================
// Tiled fp32 WMMA GEMM: C[M,N] = A[M,K] @ B[K,N], row-major.
// Block = 128 threads (4 waves, 2x2), block tile 128x128, K staged 16 deep
// through LDS; each wave computes a 64x64 tile as a 4x4 array of
// V_WMMA_F32_16X16X4_F32 accumulators (16 WMMAs per K-step).
// Requires M % 128 == 0, N % 128 == 0, K % 16 == 0.
//
// fp32 WMMA fragment layouts (ISA 7.12.2):
//   A 16x4 : VGPR v, lane l -> A[row = l%16, k = (l/16)*2 + v]
//   B 4x16 : VGPR v, lane l -> B[k  = (l/16)*2 + v, col = l%16]
//   C 16x16: VGPR j, lane l -> C[m  = j + (l/16)*8, n = l%16]
//
// LDS padding: A stride 20 floats -> the 16 b64 fragment reads of a
// half-wave hit 32 distinct banks (20r mod 64 = 4*perm(r)); B stride 136
// floats -> the +2-row offset of the upper half-wave is a +16 bank shift,
// so the two half-waves use disjoint bank ranges. Both strides are
// multiples of 4 so staging stores stay 16-byte aligned (ds_store_b128).
// ===========================================================================
#define BM 128
#define BN 128
#define BKS 16
#define LDAS 20
#define LDBS 136

__global__ __launch_bounds__(128)
void wmma_gemm_f32_tiled(const float* __restrict__ A, const float* __restrict__ B,
                         float* __restrict__ C, int K, int lda, int ldb, int ldc) {
    __shared__ float As[BM * LDAS];   // 10.0 KB
    __shared__ float Bs[BKS * LDBS];  //  8.5 KB

    const int t    = threadIdx.x;     // 0..127
    const int lane = t & 31;
    const int wid  = t >> 5;          // wave 0..3
    const int wy   = wid >> 1;        // wave row (M)
    const int wx   = wid & 1;         // wave col (N)
    const int half = lane >> 4;       // 0: K={0,1}, 1: K={2,3}
    const int l16  = lane & 15;
    const int tile_m = blockIdx.y * BM;
    const int tile_n = blockIdx.x * BN;

    v8f acc[4][4];
#pragma unroll
    for (int i = 0; i < 4; ++i)
#pragma unroll
        for (int j = 0; j < 4; ++j)
            acc[i][j] = (v8f){};

    for (int k0 = 0; k0 < K; k0 += BKS) {
        // ---- stage A(128x16) and B(16x128) into LDS, fully coalesced ----
#pragma unroll
        for (int p = 0; p < 4; ++p) {
            const int q = p * 512 + t * 4;          // 0..2047, step 4
            const int ar = q >> 4, ak = q & 15;     // A: [row][k]
            const float4 av = *(const float4*)(A + (size_t)(tile_m + ar) * lda + k0 + ak);
            *(float4*)&As[ar * LDAS + ak] = av;
            const int br = q >> 7, bc = q & 127;    // B: [k][col]
            const float4 bv = *(const float4*)(B + (size_t)(k0 + br) * ldb + tile_n + bc);
            *(float4*)&Bs[br * LDBS + bc] = bv;
        }
        __syncthreads();

        // ---- consume: 4 K-steps x 16 WMMAs ----
#pragma unroll
        for (int kk = 0; kk < BKS; kk += 4) {
            v2f fa[4], fb[4];
#pragma unroll
            for (int i = 0; i < 4; ++i)
                fa[i] = *(const v2f*)&As[(wy * 64 + i * 16 + l16) * LDAS + kk + half * 2];
#pragma unroll
            for (int j = 0; j < 4; ++j) {
                const int c = wx * 64 + j * 16 + l16;
                v2f f;
                f.x = Bs[(kk + half * 2)     * LDBS + c];
                f.y = Bs[(kk + half * 2 + 1) * LDBS + c];
                fb[j] = f;
            }
#pragma unroll
            for (int i = 0; i < 4; ++i)
#pragma unroll
                for (int j = 0; j < 4; ++j)
                    acc[i][j] = __builtin_amdgcn_wmma_f32_16x16x4_f32(
                        false, fa[i], false, fb[j], (short)0, acc[i][j], false, false);
        }
        __syncthreads();
    }

    // ---- epilogue: write the wave's 64x64 tile ----
#pragma unroll
    for (int i = 0; i < 4; ++i) {
        const int m0 = tile_m + wy * 64 + i * 16 + half * 8;
#pragma unroll
        for (int j = 0; j < 4; ++j) {
            const int n0 = tile_n + wx * 64 + j * 16 + l16;
#pragma unroll
            for (int r = 0; r < 8; ++r)
                C[(size_t)(m0 + r) * ldc + n0] = acc[i][j][r];
        }
    }
}

// ===========================================================================
// Small-N fp32 WMMA GEMM (one wave per 32x32 tile, direct from global).
// Used only for x_dbl (N = 96). Requires M,N % 32 == 0, K % 4 == 0.
// ===========================================================================
__global__ __launch_bounds__(32)
void wmma_gemm_f32_small(const float* __restrict__ A, const float* __restrict__ B,
                         float* __restrict__ C, int K, int lda, int ldb, int ldc) {
    const int tile_n = blockIdx.x * 32;
    const int tile_m = blockIdx.y * 32;
    const int lane = threadIdx.x & 31;
    const int half = lane >> 4;
    const int l16  = lane & 15;

    v8f acc00 = {}, acc01 = {}, acc10 = {}, acc11 = {};

    const float* a0 = A + (size_t)(tile_m + l16)      * lda + half * 2;
    const float* a1 = A + (size_t)(tile_m + 16 + l16) * lda + half * 2;
    const float* b0 = B + (size_t)(half * 2) * ldb + tile_n + l16;
    const float* b1 = b0 + 16;

    for (int k = 0; k < K; k += 4) {
        v2f fa0 = *(const v2f*)(a0 + k);
        v2f fa1 = *(const v2f*)(a1 + k);
        v2f fb0, fb1;
        fb0.x = b0[(size_t)k * ldb];       fb0.y = b0[(size_t)(k + 1) * ldb];
        fb1.x = b1[(size_t)k * ldb];       fb1.y = b1[(size_t)(k + 1) * ldb];
        acc00 = __builtin_amdgcn_wmma_f32_16x16x4_f32(false, fa0, false, fb0,
                                                      (short)0, acc00, false, false);
        acc01 = __builtin_amdgcn_wmma_f32_16x16x4_f32(false, fa0, false, fb1,
                                                      (short)0, acc01, false, false);
        acc10 = __builtin_amdgcn_wmma_f32_16x16x4_f32(false, fa1, false, fb0,
                                                      (short)0, acc10, false, false);
        acc11 = __builtin_amdgcn_wmma_f32_16x16x4_f32(false, fa1, false, fb1,
                                                      (short)0, acc11, false, false);
    }

    const int m0 = tile_m + half * 8;
    const int n0 = tile_n + l16;
#pragma unroll
    for (int j = 0; j < 8; ++j) {
        C[(size_t)(m0 + j)      * ldc + n0]      = acc00[j];
        C[(size_t)(m0 + j)      * ldc + n0 + 16] = acc01[j];
        C[(size_t)(m0 + 16 + j) * ldc + n0]      = acc10[j];
        C[(size_t)(m0 + 16 + j) * ldc + n0 + 16] = acc11[j];
    }
}

__device__ __forceinline__ float silu_f(float v) {
    return v * (1.0f / (1.0f + __expf(-v)));
}

// ---------------------------------------------------------------------------
// Fused: causal depthwise conv (D_CONV=4) + bias + SiLU -> xc,
//        delta = softplus(dt_vals * W_dt + b_dt)
// ---------------------------------------------------------------------------
__global__ void conv_silu_delta_kernel(const float* __restrict__ xz,
                                       const float* __restrict__ conv_w,
                                       const float* __restrict__ conv_b,
                                       const float* __restrict__ dt_vals,
                                       const float* __restrict__ W_dt,
                                       const float* __restrict__ b_dt,
                                       float* __restrict__ xc,
                                       float* __restrict__ delta) {
    const int idx = blockIdx.x * blockDim.x + threadIdx.x;
    if (idx >= BL * D_INNER) return;
    const int d  = idx & (D_INNER - 1);
    const int bl = idx >> 11;                   // D_INNER = 2^11
    const int l  = bl & (SEQ - 1);
    const int b  = bl >> 11;                    // SEQ = 2^11

    float s = conv_b[d];
#pragma unroll
    for (int j = 0; j < D_CONV; ++j) {
        const int lj = l + j - (D_CONV - 1);
        if (lj >= 0)
            s += xz[(size_t)(b * SEQ + lj) * (2 * D_INNER) + d] * conv_w[d * D_CONV + j];
    }
    xc[idx] = silu_f(s);

    const float t = dt_vals[bl] * W_dt[d] + b_dt[d];
    delta[idx] = (t > 20.0f) ? t : log1pf(__expf(t));   // softplus
}

// ---------------------------------------------------------------------------
// Selective scan, phase 1: per (b, d, chunk) compute chunk-local
//   P[n] = prod_t exp(delta_t * A[n])   (chunk decay)
//   S[n] = scan result with h_start = 0 (chunk contribution)
// Lanes are d-contiguous -> coalesced delta/xc; B_t is wave-uniform.
// ---------------------------------------------------------------------------
__global__ void scan_phase1_kernel(const float* __restrict__ delta,
                                   const float* __restrict__ xc,
                                   const float* __restrict__ xdbl,
                                   const float* __restrict__ A_log,
                                   float* __restrict__ Pbuf,
                                   float* __restrict__ Sbuf) {
    const int tid  = blockIdx.x * blockDim.x + threadIdx.x;   // 65536 total
    const int d    = tid & (D_INNER - 1);
    const int rest = tid >> 11;
    const int b    = rest & (BATCH - 1);
    const int c    = rest >> 1;

    float Av[D_STATE], h[D_STATE], P[D_STATE];
#pragma unroll
    for (int n = 0; n < D_STATE; ++n) {
        Av[n] = -__expf(A_log[d * D_STATE + n]);
        h[n] = 0.0f;
        P[n] = 1.0f;
    }

    const int l0 = c * CHLEN;
    for (int l = l0; l < l0 + CHLEN; ++l) {
        const int bl = b * SEQ + l;
        const float dlt = delta[(size_t)bl * D_INNER + d];
        const float dx  = dlt * xc[(size_t)bl * D_INNER + d];
        const float* Bt = xdbl + (size_t)bl * (DT_RANK + 2 * D_STATE) + DT_RANK;
#pragma unroll
        for (int n = 0; n < D_STATE; ++n) {
            const float a = __expf(dlt * Av[n]);
            P[n] *= a;
            h[n] = a * h[n] + dx * Bt[n];
        }
    }

    const size_t base = ((size_t)(b * NCHUNK + c) * D_STATE) * D_INNER + d;
#pragma unroll
    for (int n = 0; n < D_STATE; ++n) {
        Pbuf[base + (size_t)n * D_INNER] = P[n];
        Sbuf[base + (size_t)n * D_INNER] = h[n];
    }
}

// ---------------------------------------------------------------------------
// Phase 2: serial combine across the 16 chunks (cheap: 16 steps per row).
// ---------------------------------------------------------------------------
__global__ void scan_phase2_kernel(const float* __restrict__ Pbuf,
                                   const float* __restrict__ Sbuf,
                                   float* __restrict__ Hstart) {
    const int tid = blockIdx.x * blockDim.x + threadIdx.x;    // 4096 total
    if (tid >= BATCH * D_INNER) return;
    const int d = tid & (D_INNER - 1);
    const int b = tid >> 11;

    float h[D_STATE];
#pragma unroll
    for (int n = 0; n < D_STATE; ++n) h[n] = 0.0f;

    for (int c = 0; c < NCHUNK; ++c) {
        const size_t base = ((size_t)(b * NCHUNK + c) * D_STATE) * D_INNER + d;
#pragma unroll
        for (int n = 0; n < D_STATE; ++n) {
            Hstart[base + (size_t)n * D_INNER] = h[n];
            h[n] = Pbuf[base + (size_t)n * D_INNER] * h[n]
                 + Sbuf[base + (size_t)n * D_INNER];
        }
    }
}

// ---------------------------------------------------------------------------
// Phase 3: re-scan each chunk from its correct entry state, fuse the
// epilogue: y = (scan_y + D_param * xc) * silu(z).  ybuf aliases delta
// (each element is read then written by the same lane).
// ---------------------------------------------------------------------------
__global__ void scan_phase3_kernel(const float* __restrict__ delta,
                                   const float* __restrict__ xc,
                                   const float* __restrict__ xdbl,
                                   const float* __restrict__ A_log,
                                   const float* __restrict__ D_param,
                                   const float* __restrict__ xz,
                                   const float* __restrict__ Hstart,
                                   float* __restrict__ ybuf) {
    const int tid  = blockIdx.x * blockDim.x + threadIdx.x;
    const int d    = tid & (D_INNER - 1);
    const int rest = tid >> 11;
    const int b    = rest & (BATCH - 1);
    const int c    = rest >> 1;

    float Av[D_STATE], h[D_STATE];
    const size_t base = ((size_t)(b * NCHUNK + c) * D_STATE) * D_INNER + d;
#pragma unroll
    for (int n = 0; n < D_STATE; ++n) {
        Av[n] = -__expf(A_log[d * D_STATE + n]);
        h[n]  = Hstart[base + (size_t)n * D_INNER];
    }
    const float Dp = D_param[d];

    const int l0 = c * CHLEN;
    for (int l = l0; l < l0 + CHLEN; ++l) {
        const int bl = b * SEQ + l;
        const float dlt = delta[(size_t)bl * D_INNER + d];
        const float xv  = xc[(size_t)bl * D_INNER + d];
        const float dx  = dlt * xv;
        const float* Bt = xdbl + (size_t)bl * (DT_RANK + 2 * D_STATE) + DT_RANK;
        const float* Ct = Bt + D_STATE;
        float y = 0.0f;
#pragma unroll
        for (int n = 0; n < D_STATE; ++n) {
            const float a = __expf(dlt * Av[n]);
            h[n] = a * h[n] + dx * Bt[n];
            y += h[n] * Ct[n];
        }
        const float zv = xz[(size_t)bl * (2 * D_INNER) + D_INNER + d];
        ybuf[(size_t)bl * D_INNER + d] = (y + Dp * xv) * silu_f(zv);
    }
}

// ---------------------------------------------------------------------------
extern "C" void kernel_launch(void* const* d_in, const int* in_sizes, int n_in,
                              void* d_out, int out_size, void* d_ws, size_t ws_size,
                              hipStream_t stream) {
    const float* x       = (const float*)d_in[0];
    const float* dt_vals = (const float*)d_in[1];
    const float* W_in    = (const float*)d_in[2];
    const float* conv_w  = (const float*)d_in[3];
    const float* conv_b  = (const float*)d_in[4];
    const float* W_x     = (const float*)d_in[5];
    const float* W_dt    = (const float*)d_in[6];
    const float* b_dt    = (const float*)d_in[7];
    const float* A_log   = (const float*)d_in[8];
    const float* D_param = (const float*)d_in[9];
    const float* W_out   = (const float*)d_in[10];
    float* out = (float*)d_out;

    // Workspace carve-up (floats)
    float* ws = (float*)d_ws;
    float* xz    = ws;                                   // 4096*4096
    float* xc    = xz    + (size_t)BL * 2 * D_INNER;     // 4096*2048
    float* delta = xc    + (size_t)BL * D_INNER;         // 4096*2048 (reused as y)
    float* xdbl  = delta + (size_t)BL * D_INNER;         // 4096*96
    float* Pbuf  = xdbl  + (size_t)BL * (DT_RANK + 2 * D_STATE);
    float* Sbuf  = Pbuf  + (size_t)BATCH * NCHUNK * D_STATE * D_INNER;
    float* Hst   = Sbuf  + (size_t)BATCH * NCHUNK * D_STATE * D_INNER;

    // 1) xz = x @ W_in   (4096x1024 @ 1024x4096)
    wmma_gemm_f32_tiled<<<dim3(2 * D_INNER / BN, BL / BM), 128, 0, stream>>>(
        x, W_in, xz, D_MODEL, D_MODEL, 2 * D_INNER, 2 * D_INNER);

    // 2) causal conv + SiLU -> xc ; softplus -> delta
    conv_silu_delta_kernel<<<(BL * D_INNER + 255) / 256, 256, 0, stream>>>(
        xz, conv_w, conv_b, dt_vals, W_dt, b_dt, xc, delta);

    // 3) x_dbl = xc @ W_x  (4096x2048 @ 2048x96)
    wmma_gemm_f32_small<<<dim3((DT_RANK + 2 * D_STATE) / 32, BL / 32), 32, 0, stream>>>(
        xc, W_x, xdbl, D_INNER, D_INNER, DT_RANK + 2 * D_STATE, DT_RANK + 2 * D_STATE);

    // 4-6) chunked selective scan (3-phase linear-recurrence decomposition)
    scan_phase1_kernel<<<(BATCH * D_INNER * NCHUNK) / 256, 256, 0, stream>>>(
        delta, xc, xdbl, A_log, Pbuf, Sbuf);
    scan_phase2_kernel<<<(BATCH * D_INNER) / 256, 256, 0, stream>>>(Pbuf, Sbuf, Hst);
    scan_phase3_kernel<<<(BATCH * D_INNER * NCHUNK) / 256, 256, 0, stream>>>(
        delta, xc, xdbl, A_log, D_param, xz, Hst, /*ybuf aliases delta*/ delta);

    // 7) out = y @ W_out  (4096x2048 @ 2048x1024)
    wmma_gemm_f32_tiled<<<dim3(D_MODEL / BN, BL / BM), 128, 0, stream>>>(
        delta, W_out, out, D_INNER, D_INNER, D_MODEL, D_MODEL);
}